// SAGEConvPassMessage_xe_only_76192719831671
// MI455X (gfx1250) — compile-verified
//
#include <hip/hip_runtime.h>
#include <hip/hip_bf16.h>

typedef __attribute__((ext_vector_type(2))) float v2f;
typedef __attribute__((ext_vector_type(8))) float v8f;

#define IN_FEAT 256
#define OUT_FEAT 256
#define KSTRIDE 260   // LDS column stride (floats); 260 ≡ 4 (mod 64) => conflict-free b64 reads

// ---------------------------------------------------------------------------
// Kernel 1: zero the edge-aggregation buffer (workspace is poisoned to 0xAA).
// ---------------------------------------------------------------------------
__global__ void zero_f32_kernel(float* __restrict__ p, int n) {
    int i = blockIdx.x * blockDim.x + threadIdx.x;
    if (i < n) p[i] = 0.0f;
}

// ---------------------------------------------------------------------------
// Kernel 2: scatter-sum edge features to destination nodes.
//   he_aggr[dst_idx[e]] += he[e]   (float atomics -> global_atomic_add_f32)
// ---------------------------------------------------------------------------
__global__ void edge_scatter_kernel(const float* __restrict__ he,
                                    const long long* __restrict__ dst_idx,
                                    float* __restrict__ he_aggr,
                                    int n_edges) {
    int stride = gridDim.x * blockDim.x;
    for (int e = blockIdx.x * blockDim.x + threadIdx.x; e < n_edges; e += stride) {
        int d = (int)dst_idx[e];
        atomicAdd(&he_aggr[d], he[e]);
    }
}

// ---------------------------------------------------------------------------
// Kernel 3: out[N, 256] = [hn | he_aggr] @ W[257, 256] + b
// fp32 WMMA (V_WMMA_F32_16X16X4_F32), W column-panel staged in LDS.
//
// Grid mapping: blockIdx = rowBlock*4 + colGroup. Each 256-thread block
// (8 wave32) covers 8 row-tiles (128 rows) x one 64-col group; all 8 waves
// share one LDS-resident W panel (256 k x 64 cols, transposed, stride 260).
//
// Per wave / k-step: 1 global b64 (A fragment) + 4 ds b64 (B fragments)
// + 4 v_wmma. The 257th input column (he_aggr * W[256,:]) and the bias are
// folded into accumulator initialization; epilogue is pure NT stores.
// ---------------------------------------------------------------------------
__global__ void sage_gemm_wmma_kernel(const float* __restrict__ hn,
                                      const float* __restrict__ W,
                                      const float* __restrict__ bias,
                                      const float* __restrict__ he_aggr,
                                      float* __restrict__ out,
                                      int n_row_tiles) {
    __shared__ float sh[64 * KSTRIDE];               // 66,560 B (< 320 KB WGP LDS)

    const int colGroup = blockIdx.x & 3;
    const int rowBlock = blockIdx.x >> 2;
    const int colBase  = colGroup * 64;

    // ---- cooperative fill of the transposed W panel: sh[col*KSTRIDE + k]
    {
        const int tcol = threadIdx.x & 63;           // 0..63 within column group
        const int trow = threadIdx.x >> 6;           // 0..3
        const float* __restrict__ wcol = W + colBase + tcol;
#pragma unroll 8
        for (int k = trow; k < IN_FEAT; k += 4)
            sh[tcol * KSTRIDE + k] = wcol[(size_t)k * OUT_FEAT];
    }
    __syncthreads();

    const int lane = threadIdx.x & 31;
    const int wave = threadIdx.x >> 5;
    const int rowTile = rowBlock * 8 + wave;
    if (rowTile >= n_row_tiles) return;

    const int rowBase = rowTile * 16;
    const int l15  = lane & 15;
    const int half = lane >> 4;                      // 0: K=0,1 / rows 0-7 ; 1: K=2,3 / rows 8-15

    // ---- init accumulators: acc[c][j] = b[col] + he_aggr[row] * W[256][col]
    float ha[8];
#pragma unroll
    for (int j = 0; j < 8; ++j)
        ha[j] = he_aggr[rowBase + half * 8 + j];

    v8f acc[4];
#pragma unroll
    for (int c = 0; c < 4; ++c) {
        const int col = colBase + c * 16 + l15;
        const float bv = bias[col];
        const float wl = W[IN_FEAT * OUT_FEAT + col]; // W row 256 (edge-feature row)
#pragma unroll
        for (int j = 0; j < 8; ++j)
            acc[c][j] = __builtin_fmaf(ha[j], wl, bv);
    }

    // ---- K loop: A from global (contiguous 8B/lane), B from LDS (b64/lane)
    const float* __restrict__ arow = hn + (size_t)(rowBase + l15) * IN_FEAT + half * 2;
#pragma unroll 4
    for (int k0 = 0; k0 < IN_FEAT; k0 += 4) {
        v2f a = *(const v2f*)(arow + k0);
        const int kA = k0 + half * 2;
#pragma unroll
        for (int c = 0; c < 4; ++c) {
            v2f bm = *(const v2f*)&sh[(c * 16 + l15) * KSTRIDE + kA]; // ds_load_b64, bank-conflict-free
            acc[c] = __builtin_amdgcn_wmma_f32_16x16x4_f32(
                /*neg_a=*/false, a, /*neg_b=*/false, bm,
                /*c_mod=*/(short)0, acc[c],
                /*reuse_a=*/false, /*reuse_b=*/false);
        }
    }

    // ---- epilogue: bias/edge-term already folded in; streaming NT stores
#pragma unroll
    for (int c = 0; c < 4; ++c) {
        const int col = colBase + c * 16 + l15;
#pragma unroll
        for (int j = 0; j < 8; ++j) {
            const int row = rowBase + half * 8 + j;
            __builtin_nontemporal_store(acc[c][j], &out[(size_t)row * OUT_FEAT + col]);
        }
    }
}

// ---------------------------------------------------------------------------
// Launch wrapper. Inputs (setup_inputs order): hn, he, dst_idx(int64), W, b.
// ---------------------------------------------------------------------------
extern "C" void kernel_launch(void* const* d_in, const int* in_sizes, int n_in,
                              void* d_out, int out_size, void* d_ws, size_t ws_size,
                              hipStream_t stream) {
    const float*     hn      = (const float*)d_in[0];
    const float*     he      = (const float*)d_in[1];
    const long long* dst_idx = (const long long*)d_in[2];
    const float*     W       = (const float*)d_in[3];
    const float*     b       = (const float*)d_in[4];
    float*           out     = (float*)d_out;

    const int n_nodes = in_sizes[0] / IN_FEAT;   // 100000
    const int n_edges = in_sizes[1];             // 3200000

    float* he_aggr = (float*)d_ws;               // n_nodes floats of scratch

    // 1) zero aggregation buffer
    {
        int blocks = (n_nodes + 255) / 256;
        zero_f32_kernel<<<blocks, 256, 0, stream>>>(he_aggr, n_nodes);
    }
    // 2) scatter-add edge features
    {
        int blocks = 2048;
        edge_scatter_kernel<<<blocks, 256, 0, stream>>>(he, dst_idx, he_aggr, n_edges);
    }
    // 3) fused concat-GEMM + bias via fp32 WMMA with LDS-staged W panel
    {
        const int n_row_tiles = n_nodes / 16;                    // 6250 (exact)
        const int row_blocks  = (n_row_tiles + 7) / 8;           // 782
        const int blocks      = row_blocks * (OUT_FEAT / 64);    // 3128
        sage_gemm_wmma_kernel<<<blocks, 256, 0, stream>>>(hn, W, b, he_aggr, out, n_row_tiles);
    }
}